// NetVLAD_V2_53472342835405
// MI455X (gfx1250) — compile-verified
//
#include <hip/hip_runtime.h>
#include <hip/hip_bf16.h>

// NetVLAD on MI455X (gfx1250), fp32 WMMA path (V_WMMA_F32_16X16X4_F32)
// + async global->LDS double-buffered staging (ASYNCcnt).
// N=16, C=128, L=H*W=4096, D=512, K=64.

typedef __attribute__((ext_vector_type(2))) float v2f;
typedef __attribute__((ext_vector_type(8))) float v8f;
typedef __attribute__((ext_vector_type(4))) int   v4i;

#define NB   16
#define CC   128
#define LL   4096
#define DD   512
#define KK   64
#define EPSN 1e-12f

#if defined(__has_builtin)
#if __has_builtin(__builtin_amdgcn_global_load_async_to_lds_b32) && \
    __has_builtin(__builtin_amdgcn_global_load_async_to_lds_b128) && \
    __has_builtin(__builtin_amdgcn_s_wait_asynccnt)
#define HAS_ASYNC_LDS 1
#endif
#endif
#ifndef HAS_ASYNC_LDS
#define HAS_ASYNC_LDS 0
#endif

#define GAS __attribute__((address_space(1)))
#define LAS __attribute__((address_space(3)))

#if HAS_ASYNC_LDS
__device__ __forceinline__ void async_g2l_b32(const float* g, float* l) {
  __builtin_amdgcn_global_load_async_to_lds_b32((GAS int*)(float*)g, (LAS int*)l, 0, 0);
}
__device__ __forceinline__ void async_g2l_b128(const float* g, float* l) {
  __builtin_amdgcn_global_load_async_to_lds_b128((GAS v4i*)(float*)g, (LAS v4i*)l, 0, 0);
}
__device__ __forceinline__ void wait_async0() { __builtin_amdgcn_s_wait_asynccnt(0); }
#else
__device__ __forceinline__ void wait_async0() {}
#endif

// ---------------------------------------------------------------------------
// Kernel 0a: invnorm[n,l] = 1 / max(||x[n,:,l]||2, eps)
// ---------------------------------------------------------------------------
__global__ __launch_bounds__(256) void l2inv_kernel(const float* __restrict__ x,
                                                    float* __restrict__ inv) {
  int idx = blockIdx.x * 256 + threadIdx.x;        // < NB*LL
  int n = idx >> 12;                               // / LL
  int l = idx & (LL - 1);
  const float* p = x + (size_t)n * CC * LL + l;
  float s = 0.f;
#pragma unroll 4
  for (int c = 0; c < CC; ++c) {
    float v = p[(size_t)c * LL];
    s += v * v;
  }
  inv[idx] = 1.0f / fmaxf(sqrtf(s), EPSN);
}

// ---------------------------------------------------------------------------
// Kernel 0b: c_norm[k,:] = centroids[k,:] / max(||centroids[k,:]||2, eps)
// ---------------------------------------------------------------------------
__global__ __launch_bounds__(256) void cnorm_kernel(const float* __restrict__ cent,
                                                    float* __restrict__ cn) {
  int k = blockIdx.x;
  int tid = threadIdx.x;
  float v0 = cent[k * DD + tid];
  float v1 = cent[k * DD + 256 + tid];
  __shared__ float red[256];
  red[tid] = v0 * v0 + v1 * v1;
  __syncthreads();
  for (int s = 128; s; s >>= 1) {
    if (tid < s) red[tid] += red[tid + s];
    __syncthreads();
  }
  float inv = 1.0f / fmaxf(sqrtf(red[0]), EPSN);
  cn[k * DD + tid] = v0 * inv;
  cn[k * DD + 256 + tid] = v1 * inv;
}

// ---------------------------------------------------------------------------
// Generic 64x64 block WMMA GEMM, C[M,N] = A[M,Kd] * B, fp32 accumulate.
// MODE 0: f      = conv_w(512x128) @ x(128x4096); epilogue *= invnorm[l], += bias[d]
// MODE 1: logits = cnorm(64x512)   @ f(512x4096)
// MODE 2: vlad   = s(64x4096)      @ f(512x4096)^T   (B staged N-major, K contig)
// Fragment layouts per CDNA5 ISA (f32 16x16x4):
//   A: lane holds A[M=lane%16][K = 2*(lane/16)+{0,1}]   -> one ds_load_b64
//   B: lane holds B[K = 2*(lane/16)+{0,1}][N=lane%16]   -> one ds_load_b64 (staged [N][K])
//   D: lane holds D[M = r+8*(lane/16)][N=lane%16], r=0..7
// Staging: async global->LDS copies (ASYNCcnt), double-buffered:
//   wait asynccnt 0 -> barrier -> issue async for chunk k+1 -> WMMAs on chunk k
// ---------------------------------------------------------------------------
template <int MODE>
__global__ __launch_bounds__(256) void gemm64_wmma(const float* __restrict__ A,
                                                   const float* __restrict__ B,
                                                   float* __restrict__ Cout,
                                                   const float* __restrict__ invn,
                                                   const float* __restrict__ bias) {
  constexpr int Ntot = (MODE == 2) ? DD : LL;
  constexpr int Kd   = (MODE == 0) ? CC : (MODE == 1) ? DD : LL;
  constexpr int BM = 64, BN = 64, BK = 32;
  constexpr int LDT = BK + 4;   // row stride 144B: 16B-aligned rows, conflict-free frags
  constexpr int lda = (MODE == 0) ? CC : (MODE == 1) ? DD : LL;
  constexpr int ldb = LL;       // x and f are L-contiguous
  constexpr int tilesN = Ntot / BN;

  const int n   = blockIdx.y;
  const int m0  = (blockIdx.x / tilesN) * BM;
  const int n0  = (blockIdx.x % tilesN) * BN;
  const int tid = threadIdx.x;
  const int wave = tid >> 5;
  const int lane = tid & 31;
  const int lo = lane & 15, hi = lane >> 4;

  const float* Ap = A + (MODE == 2 ? (size_t)n * KK * LL : (size_t)0);
  const float* Bp = B + (size_t)n * ((MODE == 0) ? (size_t)CC * LL : (size_t)DD * LL);

  __shared__ __align__(16) float As[2][BM * LDT];
  __shared__ __align__(16) float Bs[2][BN * LDT];

  // each of the 8 waves owns two 16x16 output tiles of the 4x4 tile grid
  const int t0 = wave, t1 = wave + 8;
  const int tr0 = t0 >> 2, tc0 = t0 & 3;
  const int tr1 = t1 >> 2, tc1 = t1 & 3;
  v8f acc0 = {};
  v8f acc1 = {};

  auto stage = [&](int pb, int kc) {
#if HAS_ASYNC_LDS
    // A block [BM x BK], K contiguous: 16B async copies
#pragma unroll
    for (int it = 0; it < (BM * BK / 4) / 256; ++it) {
      int i = it * 256 + tid;
      int r = i >> 3, c4 = (i & 7) * 4;
      async_g2l_b128(&Ap[(size_t)(m0 + r) * lda + kc + c4], &As[pb][r * LDT + c4]);
    }
    if (MODE == 2) {
      // B source (Ntot x Kd), K contiguous: 16B async copies
#pragma unroll
      for (int it = 0; it < (BN * BK / 4) / 256; ++it) {
        int i = it * 256 + tid;
        int r = i >> 3, c4 = (i & 7) * 4;
        async_g2l_b128(&Bp[(size_t)(n0 + r) * ldb + kc + c4], &Bs[pb][r * LDT + c4]);
      }
    } else {
      // B source (Kd x Ntot), N contiguous: transpose while staging, per-lane b32
#pragma unroll
      for (int it = 0; it < (BK * BN) / 256; ++it) {
        int i = it * 256 + tid;
        int kk = i >> 6, nn = i & 63;
        async_g2l_b32(&Bp[(size_t)(kc + kk) * ldb + n0 + nn], &Bs[pb][nn * LDT + kk]);
      }
    }
#else
    for (int i = tid; i < BM * BK; i += 256) {
      int r = i >> 5, c = i & 31;
      As[pb][r * LDT + c] = Ap[(size_t)(m0 + r) * lda + kc + c];
    }
    if (MODE == 2) {
      for (int i = tid; i < BN * BK; i += 256) {
        int r = i >> 5, c = i & 31;
        Bs[pb][r * LDT + c] = Bp[(size_t)(n0 + r) * ldb + kc + c];
      }
    } else {
      for (int i = tid; i < BK * BN; i += 256) {
        int kk = i >> 6, nn = i & 63;
        Bs[pb][nn * LDT + kk] = Bp[(size_t)(kc + kk) * ldb + n0 + nn];
      }
    }
#endif
  };

  stage(0, 0);
  int pb = 0;
  for (int kc = 0; kc < Kd; kc += BK) {
    wait_async0();        // this wave's copies into buf[pb] have landed in LDS
    __syncthreads();      // everyone's copies landed; everyone done reading buf[pb^1]
    if (kc + BK < Kd) stage(pb ^ 1, kc + BK);   // overlap next chunk with WMMAs

#pragma unroll
    for (int cc = 0; cc < BK; cc += 4) {
      const int ca = cc + 2 * hi;  // K pair base for this half-wave
      v2f a0 = *(const v2f*)&As[pb][(tr0 * 16 + lo) * LDT + ca];
      v2f b0 = *(const v2f*)&Bs[pb][(tc0 * 16 + lo) * LDT + ca];
      acc0 = __builtin_amdgcn_wmma_f32_16x16x4_f32(false, a0, false, b0,
                                                   (short)0, acc0, false, false);
      v2f a1 = *(const v2f*)&As[pb][(tr1 * 16 + lo) * LDT + ca];
      v2f b1 = *(const v2f*)&Bs[pb][(tc1 * 16 + lo) * LDT + ca];
      acc1 = __builtin_amdgcn_wmma_f32_16x16x4_f32(false, a1, false, b1,
                                                   (short)0, acc1, false, false);
    }
    pb ^= 1;
  }

  // ---- epilogue ----
#pragma unroll
  for (int t = 0; t < 2; ++t) {
    const int tr = t ? tr1 : tr0;
    const int tc = t ? tc1 : tc0;
    const v8f acc = t ? acc1 : acc0;
    const int col = n0 + tc * 16 + lo;
    float inv = 1.0f;
    if (MODE == 0) inv = invn[(size_t)n * LL + col];
#pragma unroll
    for (int r = 0; r < 8; ++r) {
      const int row = m0 + tr * 16 + r + 8 * hi;
      float v = acc[r];
      if (MODE == 0) {
        v = v * inv + bias[row];
        Cout[(size_t)n * DD * LL + (size_t)row * LL + col] = v;
      } else if (MODE == 1) {
        Cout[(size_t)n * KK * LL + (size_t)row * LL + col] = v;
      } else {
        Cout[(size_t)n * KK * DD + (size_t)row * DD + col] = v;
      }
    }
  }
}

// ---------------------------------------------------------------------------
// Kernel 3: in-place softmax over L=4096 per (n,k) row; one block per row.
// ---------------------------------------------------------------------------
__global__ __launch_bounds__(256) void softmax_kernel(float* __restrict__ p) {
  const int tid = threadIdx.x;
  float* base = p + (size_t)blockIdx.x * LL;
  float v[16];
  float mx = -3.402823466e38f;
#pragma unroll
  for (int i = 0; i < 16; ++i) {
    v[i] = base[tid + i * 256];
    mx = fmaxf(mx, v[i]);
  }
  __shared__ float red[256];
  red[tid] = mx;
  __syncthreads();
  for (int s = 128; s; s >>= 1) {
    if (tid < s) red[tid] = fmaxf(red[tid], red[tid + s]);
    __syncthreads();
  }
  mx = red[0];
  __syncthreads();
  float sum = 0.f;
#pragma unroll
  for (int i = 0; i < 16; ++i) {
    v[i] = expf(v[i] - mx);
    sum += v[i];
  }
  red[tid] = sum;
  __syncthreads();
  for (int s = 128; s; s >>= 1) {
    if (tid < s) red[tid] += red[tid + s];
    __syncthreads();
  }
  float rs = 1.0f / red[0];
#pragma unroll
  for (int i = 0; i < 16; ++i) base[tid + i * 256] = v[i] * rs;
}

// ---------------------------------------------------------------------------
extern "C" void kernel_launch(void* const* d_in, const int* in_sizes, int n_in,
                              void* d_out, int out_size, void* d_ws, size_t ws_size,
                              hipStream_t stream) {
  const float* x         = (const float*)d_in[0];  // (16,128,64,64)
  const float* conv_w    = (const float*)d_in[1];  // (512,128)
  const float* conv_b    = (const float*)d_in[2];  // (512,)
  const float* centroids = (const float*)d_in[3];  // (64,512)
  float* out = (float*)d_out;                      // (16,64,512)

  char* ws = (char*)d_ws;
  float* f      = (float*)(ws);                                  // 134,217,728 B
  float* logits = (float*)(ws + (size_t)134217728);              //  16,777,216 B
  float* invn   = (float*)(ws + (size_t)150994944);              //     262,144 B
  float* cn     = (float*)(ws + (size_t)151257088);              //     131,072 B

  // per-pixel inverse norms and normalized centroids
  l2inv_kernel<<<(NB * LL) / 256, 256, 0, stream>>>(x, invn);
  cnorm_kernel<<<KK, 256, 0, stream>>>(centroids, cn);

  // f = (conv_w @ x) * invnorm + bias           : (16) x (512x4096)
  gemm64_wmma<0><<<dim3((DD / 64) * (LL / 64), NB), 256, 0, stream>>>(
      conv_w, x, f, invn, conv_b);

  // logits = cnorm @ f                          : (16) x (64x4096)
  gemm64_wmma<1><<<dim3((KK / 64) * (LL / 64), NB), 256, 0, stream>>>(
      cn, f, logits, nullptr, nullptr);

  // softmax over spatial positions (in place)
  softmax_kernel<<<NB * KK, 256, 0, stream>>>(logits);

  // vlad = s @ f^T                              : (16) x (64x512)
  gemm64_wmma<2><<<dim3((KK / 64) * (DD / 64), NB), 256, 0, stream>>>(
      logits, f, out, nullptr, nullptr);
}